// SingleUnitSmallBlock_28054726377747
// MI455X (gfx1250) — compile-verified
//
#include <hip/hip_runtime.h>
#include <hip/hip_bf16.h>

#define CH 128
#define HOPS 3

typedef __attribute__((ext_vector_type(2))) float v2f;
typedef __attribute__((ext_vector_type(8))) float v8f;

// ---- ordered float <-> uint keys for atomic max on floats -------------------
__device__ __forceinline__ unsigned enc_key(float f) {
  unsigned u = __float_as_uint(f);
  return (u & 0x80000000u) ? ~u : (u | 0x80000000u);
}
// key==0 (init) decodes to NaN -> empty bucket -> 0 (matches reference where())
__device__ __forceinline__ float dec_key(unsigned key) {
  unsigned u = (key & 0x80000000u) ? (key & 0x7fffffffu) : ~key;
  if ((u & 0x7f800000u) == 0x7f800000u) return 0.0f;  // inf/nan => empty
  return __uint_as_float(u);
}

// ---- h = relu(x @ W_in^T + b_in) -------------------------------------------
__global__ __launch_bounds__(256) void input_proj(
    const float* __restrict__ x, const float* __restrict__ W,
    const float* __restrict__ b, float* __restrict__ h, int n) {
  long long i = (long long)blockIdx.x * blockDim.x + threadIdx.x;
  int node = (int)(i >> 7), ch = (int)(i & 127);
  if (node >= n) return;
  float acc = b[ch];
  const float* xr = x + (long long)node * 10;
  const float* wr = W + ch * 10;
#pragma unroll
  for (int k = 0; k < 10; k++) acc = fmaf(xr[k], wr[k], acc);
  h[i] = fmaxf(acc, 0.0f);
}

// ---- scatter-max over edges: aggk[dst][c] = max(h[src][c]) -----------------
// attr!=null: only edges with attr==3 (polygon subgraph), also mark sink flags.
__global__ __launch_bounds__(256) void scatter_max(
    const float* __restrict__ hsrc, const int* __restrict__ src,
    const int* __restrict__ dst, const int* __restrict__ attr,
    unsigned* __restrict__ aggk, unsigned* __restrict__ flags, int nedge) {
  long long t = (long long)blockIdx.x * blockDim.x + threadIdx.x;
  int e = (int)(t >> 5);
  int lane = (int)(t & 31);
  if (e >= nedge) return;
  if (attr && attr[e] != 3) return;
  int s = src[e], d = dst[e];
  if (flags && lane == 0) flags[d] = 1u;  // idempotent store
  const float* hp = hsrc + (long long)s * CH + lane;
  unsigned* ap = aggk + (long long)d * CH + lane;
#pragma unroll
  for (int c = 0; c < 4; c++) atomicMax(ap + c * 32, enc_key(hp[c * 32]));
}

// ---- fused: out = relu(LN(dec(aggk) @ W^T + b)*g + be + resid); nf += sig*out
// One wave = 16 rows x 128 cols using V_WMMA_F32_16X16X4_F32. Block = 8 waves.
__global__ __launch_bounds__(256) void mpnn_gemm(
    const unsigned* __restrict__ aggk, const float* __restrict__ W,
    const float* __restrict__ bias, const float* __restrict__ g,
    const float* __restrict__ be, const float* __restrict__ resid,
    float* __restrict__ out, float* __restrict__ nf,
    const float* __restrict__ gate, int n) {
  __shared__ float Wt[CH * CH];  // 64KB: Wt[k][j] = W[j][k]
  for (int i = threadIdx.x; i < CH * CH; i += 256) {
    int j = i >> 7, k = i & 127;
    Wt[k * CH + j] = W[i];  // coalesced read, transposed write
  }
  __syncthreads();

  const int wave = threadIdx.x >> 5;
  const int lane = threadIdx.x & 31;
  const int lr = lane & 15;
  const int half = lane >> 4;
  const long long row_base = (long long)blockIdx.x * 128 + wave * 16;

  long long arow = row_base + lr;
  if (arow >= n) arow = n - 1;  // clamp; garbage rows never stored
  const unsigned* ap = aggk + arow * CH + 2 * half;

  v8f acc[8];
#pragma unroll
  for (int t = 0; t < 8; t++)
#pragma unroll
    for (int r = 0; r < 8; r++) acc[t][r] = 0.0f;

  for (int kk = 0; kk < CH; kk += 4) {
    v2f a;
    a.x = dec_key(ap[kk]);      // lane half 0: K=kk,kk+1 ; half 1: K=kk+2,kk+3
    a.y = dec_key(ap[kk + 1]);
    const int krow = kk + 2 * half;
#pragma unroll
    for (int t = 0; t < 8; t++) {
      v2f bb;
      bb.x = Wt[krow * CH + t * 16 + lr];
      bb.y = Wt[(krow + 1) * CH + t * 16 + lr];
      acc[t] = __builtin_amdgcn_wmma_f32_16x16x4_f32(
          false, a, false, bb, (short)0, acc[t], false, false);
    }
  }

  // per-lane channel params: ch = 16*t + lr
  float bt[8], gt[8], bet[8], sg[8];
#pragma unroll
  for (int t = 0; t < 8; t++) {
    int ch = t * 16 + lr;
    bt[t] = bias[ch];
    gt[t] = g[ch];
    bet[t] = be[ch];
    sg[t] = gate ? (1.0f / (1.0f + expf(-gate[ch]))) : 0.0f;
  }

  // bias + row statistics (row m = r + 8*half lives in VGPR r of this half)
  float s[8], q[8];
#pragma unroll
  for (int r = 0; r < 8; r++) { s[r] = 0.0f; q[r] = 0.0f; }
#pragma unroll
  for (int t = 0; t < 8; t++)
#pragma unroll
    for (int r = 0; r < 8; r++) {
      float v = acc[t][r] + bt[t];
      acc[t][r] = v;
      s[r] += v;
      q[r] += v * v;
    }
#pragma unroll
  for (int m = 1; m < 16; m <<= 1)
#pragma unroll
    for (int r = 0; r < 8; r++) {
      s[r] += __shfl_xor(s[r], m, 32);
      q[r] += __shfl_xor(q[r], m, 32);
    }
  float mu[8], rs[8];
#pragma unroll
  for (int r = 0; r < 8; r++) {
    mu[r] = s[r] * (1.0f / CH);
    float var = q[r] * (1.0f / CH) - mu[r] * mu[r];
    rs[r] = rsqrtf(var + 1e-5f);
  }

  // epilogue: LN scale/shift + residual + relu (+ gated node_features accum)
#pragma unroll
  for (int r = 0; r < 8; r++) {
    long long row = row_base + half * 8 + r;
    if (row < n) {
      const float* rp = resid + row * CH;
      float* op = out + row * CH;
      float* np = nf ? nf + row * CH : nullptr;
#pragma unroll
      for (int t = 0; t < 8; t++) {
        int ch = t * 16 + lr;
        float v = (acc[t][r] - mu[r]) * rs[r] * gt[t] + bet[t] + rp[ch];
        v = fmaxf(v, 0.0f);
        op[ch] = v;
        if (np) np[ch] += sg[t] * v;
      }
    }
  }
}

// ---- masked column-max: thread t of 256 owns output element t --------------
// sel=0: max over sink nodes of sub ; sel=1: max over !(x[:,0]>0.1) of nf
__global__ __launch_bounds__(256) void reduce_max(
    const float* __restrict__ sub, const float* __restrict__ nf,
    const unsigned* __restrict__ flags, const float* __restrict__ x,
    unsigned* __restrict__ outk, unsigned* __restrict__ found, int n) {
  int t = threadIdx.x;
  int sel = t >> 7, ch = t & 127;
  const float* buf = sel ? nf : sub;
  float m = -INFINITY;
  int any = 0;
  for (int node = blockIdx.x; node < n; node += gridDim.x) {
    bool active = sel ? !(x[(long long)node * 10] > 0.1f) : (flags[node] != 0u);
    if (active) {
      any = 1;
      m = fmaxf(m, buf[(long long)node * CH + ch]);
    }
  }
  atomicMax(&outk[t], enc_key(m));
  if (any) atomicOr(&found[sel], 1u);
}

__global__ void finalize(const unsigned* __restrict__ outk,
                         const unsigned* __restrict__ found,
                         float* __restrict__ out) {
  int t = threadIdx.x;
  unsigned key = outk[t];
  unsigned u = (key & 0x80000000u) ? (key & 0x7fffffffu) : ~key;
  float v = __uint_as_float(u);
  bool bad = ((u & 0x7f800000u) == 0x7f800000u);
  if (!found[t >> 7] || bad) v = 1e-4f;
  out[t] = v;  // [2,128]: row0=v2(sub/sinks), row1=v3(nf/remaining)
}

extern "C" void kernel_launch(void* const* d_in, const int* in_sizes, int n_in,
                              void* d_out, int out_size, void* d_ws,
                              size_t ws_size, hipStream_t stream) {
  const float* x      = (const float*)d_in[0];
  const float* W_in   = (const float*)d_in[1];
  const float* b_in   = (const float*)d_in[2];
  const float* hop_W  = (const float*)d_in[3];
  const float* hop_b  = (const float*)d_in[4];
  const float* hop_g  = (const float*)d_in[5];
  const float* hop_be = (const float*)d_in[6];
  const float* gates  = (const float*)d_in[7];
  const float* fW     = (const float*)d_in[8];
  const float* fb     = (const float*)d_in[9];
  const float* fg     = (const float*)d_in[10];
  const float* fbe    = (const float*)d_in[11];
  const int*   eidx   = (const int*)d_in[12];
  const int*   eattr  = (const int*)d_in[13];

  const int n = in_sizes[0] / 10;
  const int E = in_sizes[13];
  const int* src = eidx;
  const int* dst = eidx + E;

  // workspace layout
  float*    h     = (float*)d_ws;                         // [n,128]
  float*    nf    = h + (size_t)n * CH;                   // [n,128]
  unsigned* aggk  = (unsigned*)(nf + (size_t)n * CH);     // [n,128]
  unsigned* flags = aggk + (size_t)n * CH;                // [n]
  unsigned* outk  = flags + n;                            // [256]
  unsigned* found = outk + 256;                           // [2]

  const size_t NB = (size_t)n * CH * sizeof(float);
  hipMemsetAsync(nf, 0, NB, stream);
  hipMemsetAsync(flags, 0, (size_t)n * sizeof(unsigned), stream);
  hipMemsetAsync(outk, 0, 256 * sizeof(unsigned) + 2 * sizeof(unsigned), stream);

  input_proj<<<(int)(((long long)n * CH + 255) / 256), 256, 0, stream>>>(
      x, W_in, b_in, h, n);

  const int sblocks = (int)(((long long)E * 32 + 255) / 256);
  const int gblocks = (n + 127) / 128;
  for (int i = 0; i < HOPS; i++) {
    hipMemsetAsync(aggk, 0, NB, stream);
    scatter_max<<<sblocks, 256, 0, stream>>>(h, src, dst, nullptr, aggk,
                                             nullptr, E);
    mpnn_gemm<<<gblocks, 256, 0, stream>>>(
        aggk, hop_W + (size_t)i * CH * CH, hop_b + i * CH, hop_g + i * CH,
        hop_be + i * CH, /*resid=*/h, /*out=*/h, nf, gates + i * CH, n);
  }

  // polygon subgraph MPNN: nf -> sub (stored in h)
  hipMemsetAsync(aggk, 0, NB, stream);
  scatter_max<<<sblocks, 256, 0, stream>>>(nf, src, dst, eattr, aggk, flags, E);
  mpnn_gemm<<<gblocks, 256, 0, stream>>>(aggk, fW, fb, fg, fbe, /*resid=*/nf,
                                         /*out=*/h, nullptr, nullptr, n);

  reduce_max<<<1024, 256, 0, stream>>>(h, nf, flags, x, outk, found, n);
  finalize<<<1, 256, 0, stream>>>(outk, found, (float*)d_out);
}